// PCEN_14336600834126
// MI455X (gfx1250) — compile-verified
//
#include <hip/hip_runtime.h>
#include <hip/hip_bf16.h>

typedef __attribute__((ext_vector_type(2))) float v2f;
typedef __attribute__((ext_vector_type(8))) float v8f;

#define PCEN_EPS 1e-6f
#define LOG2E    1.44269504088896340736f

// Raw hardware transcendentals (v_exp_f32 / v_log_f32, base-2, no libm fixups).
// Safe here: every log argument is >= PCEN_EPS (positive, normal).
__device__ __forceinline__ float fexp2(float x) { return __builtin_amdgcn_exp2f(x); }
__device__ __forceinline__ float flog2(float x) { return __builtin_amdgcn_logf(x); }
__device__ __forceinline__ float fexp(float x)  { return fexp2(x * LOG2E); }
__device__ __forceinline__ float fpowk(float l2b, float k) { return fexp2(k * l2b); }

// One wave (32 lanes) handles one band f and a group of 16 batches.
// The EMA scan over T is done in chunks of 16 time-steps as
//   M_chunk(16t x 16n) = L(16x16) * X_chunk(16k x 16n) + p(t) (x) carry(n)
// using a chain of four V_WMMA_F32_16X16X4_F32 (K=16 split into 4 slices).
__global__ __launch_bounds__(32)
void pcen_wmma_scan_kernel(const float* __restrict__ x,
                           const float* __restrict__ s_log,
                           const float* __restrict__ alpha_log,
                           const float* __restrict__ delta_log,
                           const float* __restrict__ r_log,
                           float* __restrict__ out,
                           int F, int T)
{
    const int f    = blockIdx.x % F;       // band
    const int bg   = blockIdx.x / F;       // batch group of 16
    const int lane = threadIdx.x;          // wave32
    const int n    = lane & 15;            // matrix column (batch) / A row (time)
    const int h    = lane >> 4;            // lane half
    const int kb   = 2 * h;                // K pair base within a K=4 slice

    // Per-band parameters (learn_logs=True branch).
    const float s     = fexp(s_log[f]);
    const float a     = 1.0f - s;
    const float alpha = fexp(alpha_log[f]);
    const float delta = fexp(delta_log[f]);
    const float r     = fexp(r_log[f]);
    const float dr    = fexp2(r * flog2(delta));   // delta^r

    const float l2a = flog2(a);            // log2(1-s), a in (0,1)

    // A fragments: L[t][k] = (k<=t) ? s*a^(t-k) : 0.
    // A 16x4 f32 layout: lanes 0-15 hold M=0-15 (M = lane&15),
    // VGPR pair holds K = kb, kb+1 (lanes 16-31 -> K=2,3).
    v2f la[4];
#pragma unroll
    for (int j = 0; j < 4; ++j) {
        const int k0 = 4 * j + kb;
        la[j].x = (k0     <= n) ? s * fpowk(l2a, (float)(n - k0    )) : 0.0f;
        la[j].y = (k0 + 1 <= n) ? s * fpowk(l2a, (float)(n - k0 - 1)) : 0.0f;
    }
    // Carry coefficients p[t] = a^(t+1) at this lane's D rows t = v + 8h.
    float pl[8];
#pragma unroll
    for (int v = 0; v < 8; ++v)
        pl[v] = fpowk(l2a, (float)(v + 8 * h + 1));

    // This lane's (batch,band) row; T is contiguous. Both lane halves of
    // column n point at the same row.
    const size_t rowoff = ((size_t)(bg * 16 + n) * (size_t)F + (size_t)f) * (size_t)T;
    const float* __restrict__ rowp = x   + rowoff;
    float*       __restrict__ orow = out + rowoff;

    // Seed carry with x[.,.,0]: (1-s)*x0 + s*x0 == x0 == m[0].
    float carry = rowp[0];

    for (int t0 = 0; t0 < T; t0 += 16) {
        // B fragments: X_chunk is K(time) x N(batch). Lane holds column n,
        // rows K = kb, kb+1 of each K=4 slice -> consecutive in T: float2 load.
        v2f bf[4];
#pragma unroll
        for (int j = 0; j < 4; ++j)
            bf[j] = *(const v2f*)(rowp + t0 + 4 * j + kb);

        // Pointwise x for this lane's D elements: t = v + 8h, v=0..7 (contiguous).
        const float4 xlo = *(const float4*)(rowp + t0 + 8 * h);
        const float4 xhi = *(const float4*)(rowp + t0 + 8 * h + 4);

        // Prefetch next chunk (global_prefetch_b8) to cover the serial carry chain.
        if (t0 + 16 < T)
            __builtin_prefetch(rowp + t0 + 16, 0, 0);

        // C[t][n] = p[t] * carry[n]
        v8f acc;
#pragma unroll
        for (int v = 0; v < 8; ++v)
            acc[v] = pl[v] * carry;

        // M = L*X + C as a chained K-accumulation of 16x16x4 f32 WMMAs.
        acc = __builtin_amdgcn_wmma_f32_16x16x4_f32(false, la[0], false, bf[0],
                                                    (short)0, acc, false, false);
        acc = __builtin_amdgcn_wmma_f32_16x16x4_f32(false, la[1], false, bf[1],
                                                    (short)0, acc, false, false);
        acc = __builtin_amdgcn_wmma_f32_16x16x4_f32(false, la[2], false, bf[2],
                                                    (short)0, acc, false, false);
        acc = __builtin_amdgcn_wmma_f32_16x16x4_f32(false, la[3], false, bf[3],
                                                    (short)0, acc, false, false);

        // Next carry = m[15][n], held in D VGPR 7 of lane 16+n.
        carry = __shfl(acc[7], 16 + n, 32);

        // Pointwise PCEN on this lane's 8 outputs, all in base-2 transcendentals:
        //   sm = (eps+m)^(-alpha) = exp2(-alpha*log2(eps+m))
        //   o  = (x*sm+delta)^r - delta^r = exp2(r*log2(x*sm+delta)) - dr
        const float xs[8] = {xlo.x, xlo.y, xlo.z, xlo.w, xhi.x, xhi.y, xhi.z, xhi.w};
        float o[8];
#pragma unroll
        for (int v = 0; v < 8; ++v) {
            const float m  = acc[v];
            const float sm = fexp2(-alpha * flog2(m + PCEN_EPS));
            const float y  = xs[v] * sm + delta;
            o[v] = fexp2(r * flog2(y)) - dr;
        }
        *(float4*)(orow + t0 + 8 * h)     = make_float4(o[0], o[1], o[2], o[3]);
        *(float4*)(orow + t0 + 8 * h + 4) = make_float4(o[4], o[5], o[6], o[7]);
    }
}

extern "C" void kernel_launch(void* const* d_in, const int* in_sizes, int n_in,
                              void* d_out, int out_size, void* d_ws, size_t ws_size,
                              hipStream_t stream) {
    const float* x         = (const float*)d_in[0];
    const float* s_log     = (const float*)d_in[1];
    const float* alpha_log = (const float*)d_in[2];
    const float* delta_log = (const float*)d_in[3];
    const float* r_log     = (const float*)d_in[4];
    float* out             = (float*)d_out;

    const int F = in_sizes[1];             // 128 bands
    const int B = 128;                     // batch (reference setup)
    const int T = in_sizes[0] / (B * F);   // 4096 time frames

    // One wave per (band, 16-batch group): F * B/16 = 1024 single-wave blocks.
    dim3 grid((unsigned)(F * (B / 16)));
    dim3 block(32);
    pcen_wmma_scan_kernel<<<grid, block, 0, stream>>>(
        x, s_log, alpha_log, delta_log, r_log, out, F, T);
}